// EncoderLayer_hybrid1_33698313404911
// MI455X (gfx1250) — compile-verified
//
#include <hip/hip_runtime.h>
#include <hip/hip_bf16.h>
#include <math.h>

// ---------------------------------------------------------------------------
// EncoderLayer_hybrid1: quantum-circuit attention layer, restructured so the
// dominant stage (merged 8-qubit circuit over 8192 tokens) becomes a dense
// f32 GEMM [8192,512]x[512,512] executed with V_WMMA_F32_16X16X4_F32.
// B=16, S=512, E=8, H=2, d=4. Wire i <-> bit (n-1-i) (wire 0 = MSB).
// K/V staging in the attention kernel uses the Tensor Data Mover
// (tensor_load_to_lds + s_wait_tensorcnt) when the builtin is available.
// ---------------------------------------------------------------------------

typedef __attribute__((ext_vector_type(2))) float v2f;
typedef __attribute__((ext_vector_type(8))) float v8f;
typedef __attribute__((ext_vector_type(4))) unsigned int v4u;
typedef __attribute__((ext_vector_type(8))) int v8i;
typedef __attribute__((ext_vector_type(4))) int v4i;

#define NTOK   8192     // B*S
#define EDIM   8
#define DDIM   4
#define SDIM   512
#define XPAD   516      // LDS row pad (stride 4 banks -> conflict-free frag loads)

#if defined(__has_builtin)
#if __has_builtin(__builtin_amdgcn_tensor_load_to_lds) && __has_builtin(__builtin_amdgcn_s_wait_tensorcnt)
#define HAVE_TDM 1
#endif
#endif

#ifdef HAVE_TDM
// Issue one TDM 2-D tile load Global->LDS per the CDNA5 D# layout
// (CDNA5 ISA). data_size = 4 bytes. Groups 2/3 and the
// trailing group zero (<=2D tensor). Wave-uniform arguments required.
// This toolchain exposes the 6-arg builtin:
//   (uint32x4 g0, int32x8 g1, int32x4 g2, int32x4 g3, int32x8 g4, i32 cpol)
__device__ __attribute__((always_inline)) inline
void tdm_load_2d(unsigned int lds_off, const void* gptr,
                 unsigned int tensor_d0, unsigned int tensor_d1,
                 unsigned int tile_d0, unsigned int tile_d1,
                 unsigned long long stride0_elems) {
  unsigned long long ga = (unsigned long long)gptr;
  v4u g0;
  g0.x = 1u;                                        // count=1 valid descriptor
  g0.y = lds_off;                                   // lds_addr (bytes)
  g0.z = (unsigned int)ga;                          // global_addr[31:0]
  g0.w = (unsigned int)((ga >> 32) & 0x1FFFFFFu)    // global_addr[56:32]
         | 0x80000000u;                             // type=2 ("image")
  v8i g1;
  g1[0] = (int)(2u << 16);                          // data_size=2 (4B), mask=0
  g1[1] = (int)((tensor_d0 & 0xFFFFu) << 16);       // tensor_dim0[15:0]
  g1[2] = (int)((tensor_d0 >> 16) |
                ((tensor_d1 & 0xFFFFu) << 16));     // dim0[31:16] | dim1[15:0]
  g1[3] = (int)((tensor_d1 >> 16) |
                (tile_d0 << 16));                   // dim1[31:16] | tile_dim0
  g1[4] = (int)(tile_d1 & 0xFFFFu);                 // tile_dim1 (tile_dim2=0)
  g1[5] = (int)(unsigned int)(stride0_elems & 0xFFFFFFFFull);  // dim0_stride lo
  g1[6] = (int)(unsigned int)((stride0_elems >> 32) & 0xFFFFull); // stride hi
  g1[7] = 0;                                        // dim1_stride (unused)
  v4i gz; gz[0] = 0; gz[1] = 0; gz[2] = 0; gz[3] = 0;
  v8i gz8;
#pragma unroll
  for (int j = 0; j < 8; ++j) gz8[j] = 0;
  __builtin_amdgcn_tensor_load_to_lds(g0, g1, gz, gz, gz8, 0);
}
// generic->LDS offset: LDS aperture maps addr[31:0] to wave LDS offset
__device__ __attribute__((always_inline)) inline
unsigned int lds_offset_of(const void* p) {
  return (unsigned int)(unsigned long long)p;
}
#endif

// ---- complex 2x2 gate on a 16-amplitude register-resident state ------------
__device__ __attribute__((always_inline)) inline
void apply16(float* ar, float* ai, int mask,
             float g00r, float g00i, float g01r, float g01i,
             float g10r, float g10i, float g11r, float g11i) {
#pragma unroll
  for (int m = 0; m < 16; ++m) {
    if (!(m & mask)) {
      int m1 = m | mask;
      float a0r = ar[m],  a0i = ai[m];
      float a1r = ar[m1], a1i = ai[m1];
      ar[m]  = g00r*a0r - g00i*a0i + g01r*a1r - g01i*a1i;
      ai[m]  = g00r*a0i + g00i*a0r + g01r*a1i + g01i*a1r;
      ar[m1] = g10r*a0r - g10i*a0i + g11r*a1r - g11i*a1i;
      ai[m1] = g10r*a0i + g10i*a0r + g11r*a1i + g11i*a1r;
    }
  }
}

// PennyLane Rot = RZ(omega) RY(theta) RZ(phi):
//   [[ep*c, -conj(em)*s], [em*s, conj(ep)*c]],
//   ep=exp(-0.5i(omega+phi)), em=exp(0.5i(omega-phi))
__device__ __attribute__((always_inline)) inline
void rot_gate(float phi, float theta, float omega, float* g) {
  float ct = cosf(0.5f*theta), st = sinf(0.5f*theta);
  float al = 0.5f*(omega+phi), be = 0.5f*(omega-phi);
  float ca = cosf(al), sa = sinf(al), cb = cosf(be), sb = sinf(be);
  g[0] =  ca*ct; g[1] = -sa*ct;   // g00 = ep*c
  g[2] = -cb*st; g[3] =  sb*st;   // g01 = -conj(em)*s
  g[4] =  cb*st; g[5] =  sb*st;   // g10 = em*s
  g[6] =  ca*ct; g[7] =  sa*ct;   // g11 = conj(ep)*c
}

// ---------------------------------------------------------------------------
// Kernel 1: layernorm + per-head 4-qubit circuits -> V [T,E], Q/K [H,T]
// one thread per (token, head); 16-amp state lives in VGPRs.
// ---------------------------------------------------------------------------
__global__ __launch_bounds__(256) void k_heads(
    const float* __restrict__ x, const float* __restrict__ Vw,
    const float* __restrict__ Qw, const float* __restrict__ Kw,
    float* __restrict__ Vb, float* __restrict__ Qb, float* __restrict__ Kb) {
  int gid = blockIdx.x * 256 + threadIdx.x;   // 0..16383
  int t = gid >> 1, h = gid & 1;

  float xv[EDIM];
  float mean = 0.f;
#pragma unroll
  for (int e = 0; e < EDIM; ++e) { xv[e] = x[t*EDIM + e]; mean += xv[e]; }
  mean *= 0.125f;
  float var = 0.f;
#pragma unroll
  for (int e = 0; e < EDIM; ++e) { float d = xv[e]-mean; var += d*d; }
  var *= 0.125f;
  float rstd = rsqrtf(var + 1e-5f);
  float xh[DDIM];
#pragma unroll
  for (int j = 0; j < DDIM; ++j) xh[j] = (xv[h*DDIM + j] - mean) * rstd;

  // shared prefix: RX-encoding then ring CNOT, on |0000>
  float br[16], bi[16];
#pragma unroll
  for (int m = 0; m < 16; ++m) { br[m] = 0.f; bi[m] = 0.f; }
  br[0] = 1.f;
#pragma unroll
  for (int w = 0; w < 4; ++w) {
    float c = cosf(0.5f*xh[w]), s = sinf(0.5f*xh[w]);
    apply16(br, bi, 1 << (3-w), c,0.f, 0.f,-s, 0.f,-s, c,0.f);
  }
#pragma unroll
  for (int w = 0; w < 4; ++w) {
    int cm = 1 << (3-w), tm = 1 << (3-((w+1)&3));
#pragma unroll
    for (int m = 0; m < 16; ++m) {
      if ((m & cm) && !(m & tm)) {
        int m1 = m | tm;
        float tr = br[m], ti = bi[m];
        br[m] = br[m1]; bi[m] = bi[m1]; br[m1] = tr; bi[m1] = ti;
      }
    }
  }

  // --- V branch: Rot per wire, measure Z on every wire
  {
    float ar[16], ai[16];
#pragma unroll
    for (int m = 0; m < 16; ++m) { ar[m] = br[m]; ai[m] = bi[m]; }
#pragma unroll
    for (int w = 0; w < 4; ++w) {
      float g[8]; rot_gate(Vw[h*12+3*w], Vw[h*12+3*w+1], Vw[h*12+3*w+2], g);
      apply16(ar, ai, 1 << (3-w), g[0],g[1], g[2],g[3], g[4],g[5], g[6],g[7]);
    }
#pragma unroll
    for (int w = 0; w < 4; ++w) {
      int mask = 1 << (3-w);
      float z = 0.f;
#pragma unroll
      for (int m = 0; m < 16; ++m) {
        float p = ar[m]*ar[m] + ai[m]*ai[m];
        z += (m & mask) ? -p : p;
      }
      Vb[t*EDIM + h*DDIM + w] = z;
    }
  }
  // --- Q branch: Rot per wire + extra RX on wire 0, measure wire 0
  {
    float ar[16], ai[16];
#pragma unroll
    for (int m = 0; m < 16; ++m) { ar[m] = br[m]; ai[m] = bi[m]; }
#pragma unroll
    for (int w = 0; w < 4; ++w) {
      float g[8]; rot_gate(Qw[h*13+3*w], Qw[h*13+3*w+1], Qw[h*13+3*w+2], g);
      apply16(ar, ai, 1 << (3-w), g[0],g[1], g[2],g[3], g[4],g[5], g[6],g[7]);
    }
    float c = cosf(0.5f*Qw[h*13+12]), s = sinf(0.5f*Qw[h*13+12]);
    apply16(ar, ai, 8, c,0.f, 0.f,-s, 0.f,-s, c,0.f);
    float z = 0.f;
#pragma unroll
    for (int m = 0; m < 16; ++m) {
      float p = ar[m]*ar[m] + ai[m]*ai[m];
      z += (m & 8) ? -p : p;
    }
    Qb[h*NTOK + t] = z;
  }
  // --- K branch
  {
    float ar[16], ai[16];
#pragma unroll
    for (int m = 0; m < 16; ++m) { ar[m] = br[m]; ai[m] = bi[m]; }
#pragma unroll
    for (int w = 0; w < 4; ++w) {
      float g[8]; rot_gate(Kw[h*13+3*w], Kw[h*13+3*w+1], Kw[h*13+3*w+2], g);
      apply16(ar, ai, 1 << (3-w), g[0],g[1], g[2],g[3], g[4],g[5], g[6],g[7]);
    }
    float c = cosf(0.5f*Kw[h*13+12]), s = sinf(0.5f*Kw[h*13+12]);
    apply16(ar, ai, 8, c,0.f, 0.f,-s, 0.f,-s, c,0.f);
    float z = 0.f;
#pragma unroll
    for (int m = 0; m < 16; ++m) {
      float p = ar[m]*ar[m] + ai[m]*ai[m];
      z += (m & 8) ? -p : p;
    }
    Kb[h*NTOK + t] = z;
  }
}

// ---------------------------------------------------------------------------
// Kernel 2: distance-attention. One workgroup per (b,h); K,V staged in LDS
// by the Tensor Data Mover (wave 0 issues, s_wait_tensorcnt, then barrier).
// logits L_j = -scale*(q-k_j)^2, 2-pass softmax, P@V with d=4.
// ---------------------------------------------------------------------------
__global__ __launch_bounds__(512) void k_attn(
    const float* __restrict__ Qb, const float* __restrict__ Kb,
    const float* __restrict__ Vb, float* __restrict__ HO) {
  __shared__ float Kl[SDIM];
  __shared__ float Vl[SDIM][DDIM];
  int b = blockIdx.x >> 1, h = blockIdx.x & 1;
  int i = threadIdx.x;
  int base = b * SDIM;

#ifdef HAVE_TDM
  if (threadIdx.x < 32) {
    // K row: 512 contiguous f32
    tdm_load_2d(lds_offset_of(&Kl[0]), Kb + h*NTOK + base,
                /*tensor_d0=*/SDIM, /*tensor_d1=*/1,
                /*tile_d0=*/SDIM, /*tile_d1=*/1, /*stride0=*/SDIM);
    // V tile: 512 rows x 4 f32, row stride 8 f32 (head-interleaved layout)
    tdm_load_2d(lds_offset_of(&Vl[0][0]), Vb + (unsigned long long)base*EDIM + h*DDIM,
                /*tensor_d0=*/DDIM, /*tensor_d1=*/SDIM,
                /*tile_d0=*/DDIM, /*tile_d1=*/SDIM, /*stride0=*/EDIM);
    __builtin_amdgcn_s_wait_tensorcnt(0);
  }
  __syncthreads();
#else
  Kl[i] = Kb[h*NTOK + base + i];
#pragma unroll
  for (int c = 0; c < DDIM; ++c) Vl[i][c] = Vb[(base+i)*EDIM + h*DDIM + c];
  __syncthreads();
#endif

  float q = Qb[h*NTOK + base + i];
  const float scale = 0.5f;            // 1/sqrt(d)
  float dmin = 3.4e38f;
  for (int j = 0; j < SDIM; ++j) { float d = q - Kl[j]; dmin = fminf(dmin, d*d); }
  float sum = 0.f, a0 = 0.f, a1 = 0.f, a2 = 0.f, a3 = 0.f;
  for (int j = 0; j < SDIM; ++j) {
    float d = q - Kl[j];
    float e = __expf(scale * (dmin - d*d));
    sum += e;
    a0 += e*Vl[j][0]; a1 += e*Vl[j][1]; a2 += e*Vl[j][2]; a3 += e*Vl[j][3];
  }
  float inv = 1.f / sum;
  float* o = HO + (base+i)*EDIM + h*DDIM;
  o[0] = a0*inv; o[1] = a1*inv; o[2] = a2*inv; o[3] = a3*inv;
}

// ---------------------------------------------------------------------------
// Kernel 3: compose the fixed part of the merged circuit (CNOT ring + Rot
// layer over 8 wires) into a 256x256 complex unitary U (one thread per basis
// column, amplitudes in global scratch), then emit the real 512x512 GEMM
// matrix W:  y = [re(o); im(o)] = [re(s); im(s)] @ W.
//   W[m,     n] =  Ur[n,m]   W[m,     256+n] = Ui[n,m]
//   W[256+m, n] = -Ui[n,m]   W[256+m, 256+n] = Ur[n,m]
// ---------------------------------------------------------------------------
__global__ __launch_bounds__(256) void k_build_W(
    const float* __restrict__ Mw, float* __restrict__ Uc, float* __restrict__ W) {
  int col = threadIdx.x;                  // 256 threads, 1 block
  float2* a = reinterpret_cast<float2*>(Uc) + col * 256;
  for (int m = 0; m < 256; ++m) a[m] = make_float2(m == col ? 1.f : 0.f, 0.f);
  // ring CNOT: (0,1)...(7,0)
  for (int i = 0; i < 8; ++i) {
    int cm = 1 << (7-i), tm = 1 << (7-((i+1)&7));
    for (int m = 0; m < 256; ++m) {
      if ((m & cm) && !(m & tm)) {
        int m1 = m | tm;
        float2 tmp = a[m]; a[m] = a[m1]; a[m1] = tmp;
      }
    }
  }
  // Rot layer
  for (int i = 0; i < 8; ++i) {
    float g[8]; rot_gate(Mw[3*i], Mw[3*i+1], Mw[3*i+2], g);
    int mask = 1 << (7-i);
    for (int m = 0; m < 256; ++m) {
      if (!(m & mask)) {
        int m1 = m | mask;
        float2 a0 = a[m], a1 = a[m1];
        float n0r = g[0]*a0.x - g[1]*a0.y + g[2]*a1.x - g[3]*a1.y;
        float n0i = g[0]*a0.y + g[1]*a0.x + g[2]*a1.y + g[3]*a1.x;
        float n1r = g[4]*a0.x - g[5]*a0.y + g[6]*a1.x - g[7]*a1.y;
        float n1i = g[4]*a0.y + g[5]*a0.x + g[6]*a1.y + g[7]*a1.x;
        a[m]  = make_float2(n0r, n0i);
        a[m1] = make_float2(n1r, n1i);
      }
    }
  }
  // this thread owns column `col` of U -> rows col and 256+col of W
  for (int n = 0; n < 256; ++n) {
    float ur = a[n].x, ui = a[n].y;
    W[col*512       + n      ] =  ur;
    W[col*512       + 256 + n] =  ui;
    W[(256+col)*512 + n      ] = -ui;
    W[(256+col)*512 + 256 + n] =  ur;
  }
}

// ---------------------------------------------------------------------------
// Kernel 4: encoding (closed-form product state) + [16,512]x[512,512] WMMA
// GEMM tile + Z-measurement reduction + residual. 256 thr = 8 waves = 16 tok.
// Wave w owns N-tiles {w, w+8} (real cols) and {w+16, w+24} (imag cols).
// ---------------------------------------------------------------------------
__global__ __launch_bounds__(256) void k_merged(
    const float* __restrict__ HO, const float* __restrict__ W,
    const float* __restrict__ x, float* __restrict__ out) {
  __shared__ float Xl[16][XPAD];
  __shared__ float csc[16][EDIM], css[16][EDIM];
  __shared__ float zacc[16][EDIM];
  int tid  = threadIdx.x;
  int tok0 = blockIdx.x * 16;

  if (tid < 128) {
    int t = tid >> 3, e = tid & 7;
    float ang = HO[(tok0+t)*EDIM + e];
    csc[t][e] = cosf(0.5f*ang);
    css[t][e] = sinf(0.5f*ang);
    zacc[t][e] = 0.f;
  }
  __syncthreads();

  // encoding: s_m = (-i)^popcount(m) * prod_i (bit_i(m) ? sin : cos)
  {
    int t = tid >> 4;
    int mbase = (tid & 15) * 16;
#pragma unroll
    for (int mm = 0; mm < 16; ++mm) {
      int m = mbase + mm;
      float r = 1.f;
#pragma unroll
      for (int i = 0; i < 8; ++i) r *= ((m >> (7-i)) & 1) ? css[t][i] : csc[t][i];
      int pc = __popc(m) & 3;
      Xl[t][m]       = (pc == 0) ?  r : (pc == 2) ? -r : 0.f;   // re
      Xl[t][256 + m] = (pc == 1) ? -r : (pc == 3) ?  r : 0.f;   // im
    }
  }
  __syncthreads();

  int wave = tid >> 5, lane = tid & 31;
  int row   = lane & 15;               // M (token) for A-frag, N col for B-frag
  int khalf = (lane >> 4) << 1;        // lanes 16-31 hold K+2, K+3
  int n0[4];
#pragma unroll
  for (int q = 0; q < 4; ++q) n0[q] = (wave + 8*q) * 16;

  v8f acc[4];
#pragma unroll
  for (int q = 0; q < 4; ++q) {
#pragma unroll
    for (int e = 0; e < 8; ++e) acc[q][e] = 0.f;
  }

  for (int kk = 0; kk < SDIM; kk += 4) {
    int kb = kk + khalf;
    v2f a;
    a.x = Xl[row][kb];
    a.y = Xl[row][kb + 1];
#pragma unroll
    for (int q = 0; q < 4; ++q) {
      v2f bf;
      bf.x = W[kb*SDIM       + n0[q] + row];
      bf.y = W[(kb+1)*SDIM   + n0[q] + row];
      acc[q] = __builtin_amdgcn_wmma_f32_16x16x4_f32(
          false, a, false, bf, (short)0, acc[q], false, false);
    }
    if (kk + 8 < SDIM)
      __builtin_prefetch(&W[(kk+8)*SDIM + n0[0] + row], 0, 1);
  }

  // measurement: z_w = sum_n sign_w(n) * (re_n^2 + im_n^2)
#pragma unroll
  for (int p = 0; p < 2; ++p) {
    int n = n0[p] + row;                     // 0..255 (real-column index)
#pragma unroll
    for (int r8 = 0; r8 < 8; ++r8) {
      float re = acc[p][r8], im = acc[p+2][r8];
      float prob = re*re + im*im;
      int tk = r8 + ((lane >> 4) << 3);      // C layout: lanes>=16 hold M=8..15
#pragma unroll
      for (int qb = 0; qb < 8; ++qb) {
        float v = ((n >> (7-qb)) & 1) ? -prob : prob;
        atomicAdd(&zacc[tk][qb], v);
      }
    }
  }
  __syncthreads();

  if (tid < 128) {
    int t = tid >> 3, e = tid & 7;
    int tg = tok0 + t;
    out[tg*EDIM + e] = zacc[t][e] + x[tg*EDIM + e];
  }
}

// ---------------------------------------------------------------------------
extern "C" void kernel_launch(void* const* d_in, const int* in_sizes, int n_in,
                              void* d_out, int out_size, void* d_ws, size_t ws_size,
                              hipStream_t stream) {
  (void)in_sizes; (void)n_in; (void)out_size; (void)ws_size;
  const float* x  = (const float*)d_in[0];   // [16,512,8]
  const float* Vw = (const float*)d_in[1];   // [2,12]
  const float* Qw = (const float*)d_in[2];   // [2,13]
  const float* Kw = (const float*)d_in[3];   // [2,13]
  const float* Mw = (const float*)d_in[4];   // [24]
  float* out = (float*)d_out;                // [16,512,8]

  // workspace layout (floats): ~2.2 MB total
  float* ws = (float*)d_ws;
  float* Vb = ws;                 // 65536  : V  [T,E] (head-concatenated)
  float* Qb = Vb + 65536;         // 16384  : Q  [H,T]
  float* Kb = Qb + 16384;         // 16384  : K  [H,T]
  float* HO = Kb + 16384;         // 65536  : head_out [T,E]
  float* Uc = HO + 65536;         // 131072 : U column scratch (256x256 cplx)
  float* Wm = Uc + 131072;        // 262144 : real GEMM matrix W [512,512]

  k_build_W<<<dim3(1),   dim3(256), 0, stream>>>(Mw, Uc, Wm);        // indep.
  k_heads  <<<dim3(64),  dim3(256), 0, stream>>>(x, Vw, Qw, Kw, Vb, Qb, Kb);
  k_attn   <<<dim3(32),  dim3(512), 0, stream>>>(Qb, Kb, Vb, HO);
  k_merged <<<dim3(512), dim3(256), 0, stream>>>(HO, Wm, x, out);
}